// EdgeConvLayer_53652731462318
// MI455X (gfx1250) — compile-verified
//
#include <hip/hip_runtime.h>

typedef __attribute__((ext_vector_type(2))) float v2f;
typedef __attribute__((ext_vector_type(8))) float v8f;

#define IN_DIM 128
#define OUT_DIM 128
#define KNBR 16

// ---------------------------------------------------------------------------
// Prep: W3 = Um @ mW2  (Um = uW1[:,128:]),  c = ub1 + Um @ mb2
// 128x128x128 = 4 MFLOP, one block.
// ---------------------------------------------------------------------------
__global__ __launch_bounds__(256) void prep_kernel(
    const float* __restrict__ uW1, const float* __restrict__ mW2,
    const float* __restrict__ mb2, const float* __restrict__ ub1,
    float* __restrict__ W3, float* __restrict__ c)
{
    const int tid = threadIdx.x;
    for (int idx = tid; idx < 128 * 128; idx += 256) {
        const int o = idx >> 7, d = idx & 127;
        float s = 0.f;
        #pragma unroll 8
        for (int p = 0; p < 128; ++p)
            s += uW1[o * 256 + 128 + p] * mW2[p * 128 + d];
        W3[idx] = s;
    }
    if (tid < 128) {
        float s = ub1[tid];
        #pragma unroll 8
        for (int p = 0; p < 128; ++p)
            s += uW1[tid * 256 + 128 + p] * mb2[p];
        c[tid] = s;
    }
}

// ---------------------------------------------------------------------------
// Generic tall GEMM: Out[e, 0:128] = act( A1[e,:]*W1^T (+ A2[e,:]*W2^T) + bias )
// K = 128 per input, N = 128.  f32 WMMA 16x16x4, f32 accumulate (full precision).
// Block = 256 threads = 8 waves; wave owns 16 rows x 128 cols (8 acc tiles).
// Weights staged in LDS (64 KB, pad 130 floats/row vs 64 banks -> conflict-free).
// ---------------------------------------------------------------------------
template <int DUAL, int RELU>
__global__ __launch_bounds__(256) void gemm_kernel(
    const float* __restrict__ A1, const float* __restrict__ W1, int ldw1, int off1,
    const float* __restrict__ A2, const float* __restrict__ W2, int ldw2, int off2,
    const float* __restrict__ bias, float* __restrict__ Out, int E)
{
    __shared__ float wl[128 * 130];

    const int tid  = threadIdx.x;
    const int wave = tid >> 5;
    const int lane = tid & 31;
    const int l16  = lane & 15;
    const int lh   = lane >> 4;            // half-wave select (0/1)

    // stage W1 into LDS
    for (int idx = tid; idx < 128 * 128; idx += 256) {
        const int o = idx >> 7, d = idx & 127;
        wl[o * 130 + d] = W1[(size_t)o * ldw1 + off1 + d];
    }
    __syncthreads();

    const int rowbase = blockIdx.x * 128 + wave * 16;
    int arow = rowbase + l16;
    if (arow >= E) arow = E - 1;           // clamp: keep EXEC all-ones for WMMA
    const float* aptr = A1 + (size_t)arow * 128;

    v8f acc[8] = {};

    // A fragment (16x4 f32): lane m=l16, K = 4*ks + 2*lh + {0,1}
    // B fragment (4x16 f32): lane n=l16, K = 4*ks + 2*lh + {0,1}
    #pragma unroll 4
    for (int ks = 0; ks < 32; ++ks) {
        const int kk = ks * 4 + lh * 2;
        v2f a = *(const v2f*)(aptr + kk);
        #pragma unroll
        for (int nt = 0; nt < 8; ++nt) {
            v2f b = *(const v2f*)(&wl[(nt * 16 + l16) * 130 + kk]);
            acc[nt] = __builtin_amdgcn_wmma_f32_16x16x4_f32(
                false, a, false, b, (short)0, acc[nt], false, false);
        }
    }

    if (DUAL) {
        __syncthreads();                   // everyone done reading W1
        for (int idx = tid; idx < 128 * 128; idx += 256) {
            const int o = idx >> 7, d = idx & 127;
            wl[o * 130 + d] = W2[(size_t)o * ldw2 + off2 + d];
        }
        __syncthreads();

        const float* aptr2 = A2 + (size_t)arow * 128;
        #pragma unroll 4
        for (int ks = 0; ks < 32; ++ks) {
            const int kk = ks * 4 + lh * 2;
            v2f a = *(const v2f*)(aptr2 + kk);
            #pragma unroll
            for (int nt = 0; nt < 8; ++nt) {
                v2f b = *(const v2f*)(&wl[(nt * 16 + l16) * 130 + kk]);
                acc[nt] = __builtin_amdgcn_wmma_f32_16x16x4_f32(
                    false, a, false, b, (short)0, acc[nt], false, false);
            }
        }
    }

    // C/D layout: VGPR i, lanes 0-15 -> row i, lanes 16-31 -> row i+8; col = l16.
    #pragma unroll
    for (int nt = 0; nt < 8; ++nt) {
        const int col = nt * 16 + l16;
        const float bv = bias ? bias[col] : 0.f;
        #pragma unroll
        for (int i = 0; i < 8; ++i) {
            const int row = rowbase + i + lh * 8;
            float v = acc[nt][i] + bv;
            if (RELU) v = fmaxf(v, 0.f);
            if (row < E) Out[(size_t)row * 128 + col] = v;
        }
    }
}

// ---------------------------------------------------------------------------
// Gather pass: hbar[e,:] = (1/16) * sum_k relu(S[e,:] + P[adj[e,k],:] + mb1)
// One wave per edge, one float4 per lane (32*4 = 128 dims). P is L2-resident
// (51 MB vs 192 MB L2), so this is an L2-bandwidth pass: load all 16 indices
// first (uniform -> scalar loads), then issue all 16 gathered b128 row loads
// back-to-back for full memory-level parallelism, then accumulate.
// hbar may alias S (element [e,d] reads only S[e,d] before writing it).
// ---------------------------------------------------------------------------
__global__ __launch_bounds__(256) void gather_kernel(
    const float* __restrict__ S, const float* __restrict__ P,
    const long long* __restrict__ adj, const float* __restrict__ mb1,
    float* __restrict__ hbar, int E)
{
    const int lane = threadIdx.x & 31;
    const int wave = threadIdx.x >> 5;
    const int e = blockIdx.x * 8 + wave;
    if (e >= E) return;

    // all 16 neighbor indices up front (uniform per wave)
    const long long* arow = adj + (size_t)e * KNBR;
    long long idx[KNBR];
    #pragma unroll
    for (int k = 0; k < KNBR; ++k) idx[k] = arow[k];

    const float4 b = ((const float4*)mb1)[lane];
    const float4 s = ((const float4*)(S + (size_t)e * 128))[lane];

    // all 16 gathered rows in flight (64 VGPRs/lane of payload)
    float4 p[KNBR];
    #pragma unroll
    for (int k = 0; k < KNBR; ++k)
        p[k] = ((const float4*)(P + (size_t)idx[k] * 128))[lane];

    float ax = 0.f, ay = 0.f, az = 0.f, aw = 0.f;
    #pragma unroll
    for (int k = 0; k < KNBR; ++k) {
        ax += fmaxf(s.x + p[k].x + b.x, 0.f);
        ay += fmaxf(s.y + p[k].y + b.y, 0.f);
        az += fmaxf(s.z + p[k].z + b.z, 0.f);
        aw += fmaxf(s.w + p[k].w + b.w, 0.f);
    }
    float4 r;
    r.x = ax * (1.f / KNBR); r.y = ay * (1.f / KNBR);
    r.z = az * (1.f / KNBR); r.w = aw * (1.f / KNBR);
    ((float4*)(hbar + (size_t)e * 128))[lane] = r;
}

// ---------------------------------------------------------------------------
// Driver
// ---------------------------------------------------------------------------
extern "C" void kernel_launch(void* const* d_in, const int* in_sizes, int n_in,
                              void* d_out, int out_size, void* d_ws, size_t ws_size,
                              hipStream_t stream)
{
    const float*     X    = (const float*)d_in[0];      // (E,128)
    const long long* adj  = (const long long*)d_in[1];  // (E,16) int64
    const float*     mW1  = (const float*)d_in[2];      // (128,256)
    const float*     mb1  = (const float*)d_in[3];      // (128,)
    const float*     mW2  = (const float*)d_in[4];      // (128,128)
    const float*     mb2  = (const float*)d_in[5];      // (128,)
    const float*     uW1  = (const float*)d_in[6];      // (128,256)
    const float*     ub1  = (const float*)d_in[7];      // (128,)
    const float*     uW2  = (const float*)d_in[8];      // (128,128)
    const float*     ub2  = (const float*)d_in[9];      // (128,)
    float*           out  = (float*)d_out;

    const int E = in_sizes[0] / IN_DIM;
    const size_t mat = (size_t)E * IN_DIM;              // floats per (E,128)

    float* S   = (float*)d_ws;                          // S, later hbar (in-place)
    float* P   = S + mat;                               // P, later u (reuse)
    float* W3  = P + mat;                               // 128*128
    float* c   = W3 + 128 * 128;                        // 128

    const int gemm_blocks = (E + 127) / 128;

    // 0) fold mW2/mb2 into the update path
    prep_kernel<<<1, 256, 0, stream>>>(uW1, mW2, mb2, ub1, W3, c);

    // 1) S = X @ Wi^T   (Wi = mW1[:, :128], row stride 256)
    gemm_kernel<0, 0><<<gemm_blocks, 256, 0, stream>>>(
        X, mW1, 256, 0, nullptr, nullptr, 0, 0, nullptr, S, E);

    // 2) P = X @ Wj^T   (Wj = mW1[:, 128:])
    gemm_kernel<0, 0><<<gemm_blocks, 256, 0, stream>>>(
        X, mW1, 256, 128, nullptr, nullptr, 0, 0, nullptr, P, E);

    // 3) hbar = mean_k relu(S + P[adj] + mb1)   (writes over S)
    gather_kernel<<<(E + 7) / 8, 256, 0, stream>>>(S, P, adj, mb1, S, E);

    // 4) u = relu(X @ Ui^T + hbar @ W3^T + c)   (writes over P)
    gemm_kernel<1, 1><<<gemm_blocks, 256, 0, stream>>>(
        X, uW1, 256, 0, S, W3, 128, 0, c, P, E);

    // 5) out = u @ uW2^T + ub2
    gemm_kernel<0, 0><<<gemm_blocks, 256, 0, stream>>>(
        P, uW2, 128, 0, nullptr, nullptr, 0, 0, ub2, out, E);
}